// LocalSpatialEncoding_24670292148535
// MI455X (gfx1250) — compile-verified
//
#include <hip/hip_runtime.h>

#define NPTS   8192
#define NB     2
#define DCH    8
#define KNNK   16
#define BN_EPS 1e-6f

typedef __attribute__((ext_vector_type(2))) float v2f;
typedef __attribute__((ext_vector_type(8))) float v8f;

// ---------------- kernel 0: |p|^2 per point + zero BN-stat accumulators ----------------
__global__ void sq_init_kernel(const float* __restrict__ coords,
                               float* __restrict__ sq,
                               float* __restrict__ stats) {
  int i = blockIdx.x * blockDim.x + threadIdx.x;
  if (i < NB * NPTS) {
    const float* p = coords + (size_t)i * 3;
    sq[i] = p[0] * p[0] + p[1] * p[1] + p[2] * p[2];
  }
  if (blockIdx.x == 0 && threadIdx.x < 2 * DCH) stats[threadIdx.x] = 0.0f;
}

// ---------------- kernel 1: KNN via WMMA f32 16x16x4 distance tiles ----------------
// One wave per 32 query rows. For a 16-col tile:
//   A row i = [x_i, y_i, z_i, 1]  (16x4)
//   B col j = [-2x_j, -2y_j, -2z_j, sq_j]  (4x16)
//   A*B = sq_j - 2 x_i.x_j ; add sq_i after the LDS transpose -> d2(i,j).
// A layout (ISA 32-bit 16x4): lanes 0-15 -> {K0,K1} in {v0,v1}, lanes 16-31 -> {K2,K3}.
// B layout assumed symmetric (rows striped across lanes per VGPR, +2 for hi half).
__global__ __launch_bounds__(32) void knn_kernel(const float* __restrict__ coords,
                                                 const float* __restrict__ sq,
                                                 float* __restrict__ knnd,
                                                 int* __restrict__ knni) {
  const int b    = blockIdx.y;
  const int row0 = blockIdx.x * 32;
  const int lane = threadIdx.x;
  const int m    = lane & 15;
  const bool hi  = lane >= 16;

  const float* cb  = coords + (size_t)b * NPTS * 3;
  const float* sqb = sq + (size_t)b * NPTS;

  // A matrices for rows [row0,row0+16) and [row0+16,row0+32)
  v2f a0, a1;
  {
    const float* p0 = cb + (size_t)(row0 + m) * 3;
    const float* p1 = cb + (size_t)(row0 + 16 + m) * 3;
    if (!hi) { a0.x = p0[0]; a0.y = p0[1]; a1.x = p1[0]; a1.y = p1[1]; }
    else     { a0.x = p0[2]; a0.y = 1.0f;  a1.x = p1[2]; a1.y = 1.0f; }
  }
  const float sq_own = sqb[row0 + lane];

  // per-lane sorted (ascending) top-16 list for row (row0+lane)
  float dl[KNNK]; int il[KNNK];
  #pragma unroll
  for (int j = 0; j < KNNK; ++j) { dl[j] = 3.0e38f; il[j] = 0; }

  __shared__ float tile[32 * 17];  // pad 16 -> 17 floats/row: bank-conflict-free row reads

  for (int t = 0; t < NPTS / 16; ++t) {
    const int col0 = t * 16;
    v2f bm;
    {
      const float* q = cb + (size_t)(col0 + m) * 3;
      if (!hi) { bm.x = -2.0f * q[0]; bm.y = -2.0f * q[1]; }
      else     { bm.x = -2.0f * q[2]; bm.y = sqb[col0 + m]; }
    }
    v8f cz = {0.f, 0.f, 0.f, 0.f, 0.f, 0.f, 0.f, 0.f};
    v8f c0 = __builtin_amdgcn_wmma_f32_16x16x4_f32(false, a0, false, bm, (short)0, cz, false, false);
    v8f c1 = __builtin_amdgcn_wmma_f32_16x16x4_f32(false, a1, false, bm, (short)0, cz, false, false);

    // C/D layout: VGPR v -> row (v + 8*hiHalf), col = lane&15. Transpose through LDS.
    const int half = hi ? 8 : 0;
    #pragma unroll
    for (int v = 0; v < 8; ++v) {
      tile[(half + v) * 17 + m]      = c0[v];
      tile[(16 + half + v) * 17 + m] = c1[v];
    }
    __syncthreads();  // single-wave WG: S_NOP + compiler DScnt wait

    // Lane r owns row r: filter 16 candidates into its sorted top-16.
    #pragma unroll
    for (int j = 0; j < 16; ++j) {
      float cand = tile[lane * 17 + j] + sq_own;
      int ci = col0 + j;
      if (cand < dl[KNNK - 1]) {   // strict: equal-to-worst keeps earlier index (top_k stable)
        dl[KNNK - 1] = cand; il[KNNK - 1] = ci;
        #pragma unroll
        for (int u = KNNK - 1; u > 0; --u) {
          if (dl[u] < dl[u - 1]) { // no swap on ties -> later index stays later (stable)
            float td = dl[u]; dl[u] = dl[u - 1]; dl[u - 1] = td;
            int   ti = il[u]; il[u] = il[u - 1]; il[u - 1] = ti;
          }
        }
      }
    }
    __syncthreads();
  }

  float* dd = knnd + ((size_t)b * NPTS + row0 + lane) * KNNK;
  int*   ii = knni + ((size_t)b * NPTS + row0 + lane) * KNNK;
  #pragma unroll
  for (int j = 0; j < KNNK; ++j) { dd[j] = dl[j]; ii[j] = il[j]; }
}

// ---------------- shared: rebuild feat10 + 1x1 conv for one (b,n,k) ----------------
__device__ __forceinline__ void compute_y(const float* __restrict__ coords,
                                          const float* __restrict__ knnd,
                                          const int* __restrict__ knni,
                                          const float* __restrict__ w,
                                          const float* __restrict__ bias,
                                          int b, int n, int k, float* y) {
  size_t base = ((size_t)b * NPTS + n) * KNNK + k;
  int   j    = knni[base];
  float dist = knnd[base];
  const float* pc = coords + ((size_t)b * NPTS + n) * 3;
  const float* pn = coords + ((size_t)b * NPTS + j) * 3;
  float f[10];
  f[0] = pc[0]; f[1] = pc[1]; f[2] = pc[2];
  f[3] = pn[0]; f[4] = pn[1]; f[5] = pn[2];
  f[6] = f[0] - f[3]; f[7] = f[1] - f[4]; f[8] = f[2] - f[5];
  f[9] = dist;
  #pragma unroll
  for (int d = 0; d < DCH; ++d) {
    float acc = bias[d];
    #pragma unroll
    for (int c = 0; c < 10; ++c) acc = fmaf(w[d * 10 + c], f[c], acc);
    y[d] = acc;
  }
}

// ---------------- kernel 2: BN training stats (sum, sumsq per channel) ----------------
__global__ __launch_bounds__(256) void stats_kernel(const float* __restrict__ coords,
                                                    const float* __restrict__ knnd,
                                                    const int* __restrict__ knni,
                                                    const float* __restrict__ w,
                                                    const float* __restrict__ bias,
                                                    float* __restrict__ stats) {
  int i = blockIdx.x * blockDim.x + threadIdx.x;   // 0 .. NB*NPTS*KNNK-1 (exact)
  int b = i / (NPTS * KNNK);
  int r = i % (NPTS * KNNK);
  int n = r / KNNK, k = r % KNNK;

  float y[DCH];
  compute_y(coords, knnd, knni, w, bias, b, n, k, y);

  __shared__ float red[2 * DCH];
  if (threadIdx.x < 2 * DCH) red[threadIdx.x] = 0.0f;
  __syncthreads();

  #pragma unroll
  for (int d = 0; d < DCH; ++d) {
    float s = y[d], s2 = y[d] * y[d];
    #pragma unroll
    for (int off = 16; off > 0; off >>= 1) {   // wave32 reduction
      s  += __shfl_xor(s,  off, 32);
      s2 += __shfl_xor(s2, off, 32);
    }
    if ((threadIdx.x & 31) == 0) {
      atomicAdd(&red[d], s);
      atomicAdd(&red[DCH + d], s2);
    }
  }
  __syncthreads();
  if (threadIdx.x < 2 * DCH) atomicAdd(&stats[threadIdx.x], red[threadIdx.x]);
}

// ---------------- kernel 3: BN apply + ReLU + concat repeated features ----------------
__global__ __launch_bounds__(256) void out_kernel(const float* __restrict__ coords,
                                                  const float* __restrict__ feats,
                                                  const float* __restrict__ knnd,
                                                  const int* __restrict__ knni,
                                                  const float* __restrict__ w,
                                                  const float* __restrict__ bias,
                                                  const float* __restrict__ gamma,
                                                  const float* __restrict__ beta,
                                                  const float* __restrict__ stats,
                                                  float* __restrict__ out) {
  int i = blockIdx.x * blockDim.x + threadIdx.x;
  int b = i / (NPTS * KNNK);
  int r = i % (NPTS * KNNK);
  int n = r / KNNK, k = r % KNNK;

  float y[DCH];
  compute_y(coords, knnd, knni, w, bias, b, n, k, y);

  const float inv = 1.0f / (float)(NB * NPTS * KNNK);
  #pragma unroll
  for (int d = 0; d < DCH; ++d) {
    float mean = stats[d] * inv;
    float var  = stats[DCH + d] * inv - mean * mean;   // biased var = E[y^2]-E[y]^2
    float v = (y[d] - mean) * rsqrtf(var + BN_EPS) * gamma[d] + beta[d];
    v = fmaxf(v, 0.0f);
    out[(((size_t)b * 2 * DCH + d) * NPTS + n) * KNNK + k] = v;
    out[(((size_t)b * 2 * DCH + DCH + d) * NPTS + n) * KNNK + k] =
        feats[((size_t)b * DCH + d) * NPTS + n];
  }
}

extern "C" void kernel_launch(void* const* d_in, const int* in_sizes, int n_in,
                              void* d_out, int out_size, void* d_ws, size_t ws_size,
                              hipStream_t stream) {
  const float* coords = (const float*)d_in[0];  // (B,N,3)
  const float* feats  = (const float*)d_in[1];  // (B,d,N,1)
  const float* w      = (const float*)d_in[2];  // (d,10)
  const float* bias   = (const float*)d_in[3];  // (d,)
  const float* gamma  = (const float*)d_in[4];  // (d,)
  const float* beta   = (const float*)d_in[5];  // (d,)
  float* out = (float*)d_out;                   // (B,2d,N,K)

  // workspace layout (~2.2 MB)
  float* sq    = (float*)d_ws;                              // NB*NPTS
  float* knnd  = sq + (size_t)NB * NPTS;                    // NB*NPTS*KNNK
  int*   knni  = (int*)(knnd + (size_t)NB * NPTS * KNNK);   // NB*NPTS*KNNK
  float* stats = (float*)(knni + (size_t)NB * NPTS * KNNK); // 2*DCH

  sq_init_kernel<<<(NB * NPTS + 255) / 256, 256, 0, stream>>>(coords, sq, stats);

  dim3 g(NPTS / 32, NB);
  knn_kernel<<<g, 32, 0, stream>>>(coords, sq, knnd, knni);

  int total = NB * NPTS * KNNK;  // 262144
  stats_kernel<<<total / 256, 256, 0, stream>>>(coords, knnd, knni, w, bias, stats);
  out_kernel<<<total / 256, 256, 0, stream>>>(coords, feats, knnd, knni, w, bias,
                                              gamma, beta, stats, out);
}